// GCN_3layer_6889127543167
// MI455X (gfx1250) — compile-verified
//
#include <hip/hip_runtime.h>
#include <stdint.h>

// ---------------------------------------------------------------------------
// GCN 3-layer inference for MI455X (gfx1250, wave32).
//
// Roofline: HBM-bound (~200MB moved vs ~1.8 GFLOP at 23.3 TB/s ~ 9us).  GEMMs
// use native fp32 WMMA (V_WMMA_F32_16X16X4_F32): numerics match the fp32
// reference and matrix-core throughput is irrelevant here, so bf16/fp8
// downconversion buys nothing.  Sparse aggregation uses lane-per-column edge
// parallelism: one coalesced 128B row gather + 32 global_atomic_add_f32 per
// edge; the 19.2MB feature matrix is L2-resident (192MB L2).
//
// v2 change: W is staged into LDS pre-swizzled into WMMA B-fragment order
// (K-pairs contiguous, pair-row stride padded to 96 floats) so each lane's B
// fragment is ONE aligned ds_load_b64 into an even VGPR pair — removes the
// ds_load_2addr_b32 + v_mov packing seen in v1, and the two half-wave groups
// read disjoint LDS bank halves (conflict-free).
// ---------------------------------------------------------------------------

typedef float v2f __attribute__((ext_vector_type(2)));
typedef float v8f __attribute__((ext_vector_type(8)));

#define HID   32
#define FIN   128
#define TPB   256
#define BPAD  96          // pair-row stride in floats: 64 data + 32 pad

// ---------------- degree / normalization ----------------

__global__ void deg_init_kernel(float* __restrict__ deg, int n) {
  int i = blockIdx.x * blockDim.x + threadIdx.x;
  if (i < n) deg[i] = 1.0f;                       // self-loop contributes 1
}

__global__ void deg_scatter_kernel(const long long* __restrict__ ei,
                                   float* __restrict__ deg, int nE) {
  int e = blockIdx.x * blockDim.x + threadIdx.x;
  if (e < nE) atomicAdd(&deg[(int)ei[(long long)nE + e]], 1.0f);
}

__global__ void dinv_kernel(const float* __restrict__ deg,
                            float* __restrict__ dis, int n) {
  int i = blockIdx.x * blockDim.x + threadIdx.x;
  if (i < n) dis[i] = rsqrtf(fmaxf(deg[i], 1.0f));
}

// ---------------- WMMA GEMM:  C[n x 32] = (relu?)A[n x K] @ W[K x 32] -------
// One wave -> 16 rows x 32 cols (two 16x16 f32 accumulators).
// Block of 256 threads (8 waves) -> 128 rows; W staged in LDS in fragment
// order: element W[k][n] lives at sW[(k>>1)*BPAD + n*2 + (k&1)].

template <bool RELU>
__global__ void __launch_bounds__(TPB)
gemm_wmma_kernel(const float* __restrict__ A,   // [n x K]
                 const float* __restrict__ W,   // [K x 32]
                 float* __restrict__ C,         // [n x 32]
                 int n, int K) {
  __shared__ float sW[(FIN / 2) * BPAD];         // 24KB max (K<=128)
  const int tid = threadIdx.x;
  for (int i = tid; i < K * HID; i += TPB) {
    int k  = i >> 5;
    int nn = i & 31;
    sW[(k >> 1) * BPAD + nn * 2 + (k & 1)] = W[i];
  }
  __syncthreads();

  const int lane = tid & 31;
  const int wave = tid >> 5;
  const int r0   = blockIdx.x * 128 + wave * 16;

  // A fragment layout (16x4 f32): lanes 0-15 rows M=0..15 K={0,1};
  //                               lanes 16-31 rows M=0..15 K={2,3}.
  const int m  = lane & 15;
  const int g  = lane >> 4;                      // half-wave group (0/1)
  const int kg = g * 2;                          // K offset within 4-step
  const int nn = lane & 15;                      // B/C column within tile

  int row = r0 + m;
  if (row >= n) row = n - 1;                     // clamp: EXEC stays all-1s
  const float* __restrict__ arow = A + (long long)row * K;

  // Per-lane LDS base for B fragments: pair-row (k0/2 + g), column nn.
  const float* __restrict__ bb0 = &sW[g * BPAD + nn * 2];        // N tile 0
  const float* __restrict__ bb1 = &sW[g * BPAD + (nn + 16) * 2]; // N tile 1

  v8f acc0 = {};                                 // cols 0..15
  v8f acc1 = {};                                 // cols 16..31

  for (int k0 = 0; k0 < K; k0 += 4) {
    float a0 = arow[k0 + kg];
    float a1 = arow[k0 + kg + 1];
    if (RELU) { a0 = fmaxf(a0, 0.0f); a1 = fmaxf(a1, 0.0f); }
    v2f a; a.x = a0; a.y = a1;

    const int poff = (k0 >> 1) * BPAD;           // pair-row k0/2
    v2f b0 = *(const v2f*)(bb0 + poff);          // one ds_load_b64
    v2f b1 = *(const v2f*)(bb1 + poff);          // one ds_load_b64

    acc0 = __builtin_amdgcn_wmma_f32_16x16x4_f32(false, a, false, b0,
                                                 (short)0, acc0, false, false);
    acc1 = __builtin_amdgcn_wmma_f32_16x16x4_f32(false, a, false, b1,
                                                 (short)0, acc1, false, false);
  }

  // C/D layout: VGPR v -> M=v (lanes 0-15) / M=v+8 (lanes 16-31); N=lane&15.
  const int mbase = (lane < 16) ? 0 : 8;
#pragma unroll
  for (int v = 0; v < 8; ++v) {
    int rr = r0 + mbase + v;
    if (rr < n) {
      C[(long long)rr * HID + nn]      = acc0[v];
      C[(long long)rr * HID + nn + 16] = acc1[v];
    }
  }
}

// ---------------- aggregation: self-loop init + edge scatter ----------------

__global__ void agg_init_kernel(const float* __restrict__ H,
                                const float* __restrict__ dis,
                                const float* __restrict__ bias,
                                float* __restrict__ Aout, int n) {
  long long i = (long long)blockIdx.x * blockDim.x + threadIdx.x;
  if (i < (long long)n * HID) {
    int node = (int)(i >> 5);
    int c    = (int)(i & 31);
    float d  = dis[node];
    Aout[i]  = H[i] * d * d + bias[c];           // self-loop norm = d^2
  }
}

__global__ void agg_scatter_kernel(const long long* __restrict__ ei,
                                   const float* __restrict__ H,
                                   const float* __restrict__ dis,
                                   float* __restrict__ Aout, int nE) {
  long long i = (long long)blockIdx.x * blockDim.x + threadIdx.x;
  if (i < (long long)nE * HID) {
    int e = (int)(i >> 5);
    int c = (int)(i & 31);
    int s = (int)ei[e];
    int d = (int)ei[(long long)nE + e];
    float norm = dis[s] * dis[d];
    atomicAdd(&Aout[(long long)d * HID + c], H[(long long)s * HID + c] * norm);
  }
}

// ---------------- head: relu -> 32x1 matvec -> self-loop+bias ---------------

__global__ void head_kernel(const float* __restrict__ A2,
                            const float* __restrict__ dis,
                            const float* __restrict__ W3,
                            const float* __restrict__ b3,
                            float* __restrict__ t,
                            float* __restrict__ out, int n) {
  int i = blockIdx.x * blockDim.x + threadIdx.x;
  if (i < n) {
    const float* __restrict__ row = A2 + (long long)i * HID;
    float s = 0.0f;
#pragma unroll
    for (int c = 0; c < HID; ++c) s += fmaxf(row[c], 0.0f) * W3[c];
    t[i] = s;
    float d = dis[i];
    out[i] = s * d * d + b3[0];                  // self-loop + bias
  }
}

__global__ void out_scatter_kernel(const long long* __restrict__ ei,
                                   const float* __restrict__ t,
                                   const float* __restrict__ dis,
                                   float* __restrict__ out, int nE) {
  int e = blockIdx.x * blockDim.x + threadIdx.x;
  if (e < nE) {
    int s = (int)ei[e];
    int d = (int)ei[(long long)nE + e];
    atomicAdd(&out[d], t[s] * dis[s] * dis[d]);
  }
}

__global__ void sigmoid_kernel(float* __restrict__ out, int n) {
  int i = blockIdx.x * blockDim.x + threadIdx.x;
  if (i < n) out[i] = 1.0f / (1.0f + __expf(-out[i]));
}

// ---------------------------------------------------------------------------

static inline int cdiv_ll(long long a, long long b) { return (int)((a + b - 1) / b); }

extern "C" void kernel_launch(void* const* d_in, const int* in_sizes, int n_in,
                              void* d_out, int out_size, void* d_ws, size_t ws_size,
                              hipStream_t stream) {
  const float*     x  = (const float*)d_in[0];
  const long long* ei = (const long long*)d_in[1];   // reference dtype: int64
  const float*     W1 = (const float*)d_in[2];
  const float*     b1 = (const float*)d_in[3];
  const float*     W2 = (const float*)d_in[4];
  const float*     b2 = (const float*)d_in[5];
  const float*     W3 = (const float*)d_in[6];
  const float*     b3 = (const float*)d_in[7];
  float* out = (float*)d_out;

  const int n  = in_sizes[0] / FIN;
  const int nE = in_sizes[1] / 2;

  // workspace partition (floats): deg[n] dis[n] H[n*32] A[n*32] t[n]  ~40MB
  float* ws  = (float*)d_ws;
  float* deg = ws;
  float* dis = deg + n;
  float* H   = dis + n;
  float* Ab  = H + (long long)n * HID;
  float* t   = Ab + (long long)n * HID;

  // normalization
  deg_init_kernel<<<cdiv_ll(n, TPB), TPB, 0, stream>>>(deg, n);
  deg_scatter_kernel<<<cdiv_ll(nE, TPB), TPB, 0, stream>>>(ei, deg, nE);
  dinv_kernel<<<cdiv_ll(n, TPB), TPB, 0, stream>>>(deg, dis, n);

  const int gemm_grid = cdiv_ll(n, 128);
  const int aggN      = cdiv_ll((long long)n  * HID, TPB);
  const int aggE      = cdiv_ll((long long)nE * HID, TPB);

  // layer 1: H = X @ W1 ; A = aggregate(H) + b1
  gemm_wmma_kernel<false><<<gemm_grid, TPB, 0, stream>>>(x, W1, H, n, FIN);
  agg_init_kernel<<<aggN, TPB, 0, stream>>>(H, dis, b1, Ab, n);
  agg_scatter_kernel<<<aggE, TPB, 0, stream>>>(ei, H, dis, Ab, nE);

  // layer 2: H = relu(A) @ W2 ; A = aggregate(H) + b2   (ReLU folded in load)
  gemm_wmma_kernel<true><<<gemm_grid, TPB, 0, stream>>>(Ab, W2, H, n, HID);
  agg_init_kernel<<<aggN, TPB, 0, stream>>>(H, dis, b2, Ab, n);
  agg_scatter_kernel<<<aggE, TPB, 0, stream>>>(ei, H, dis, Ab, nE);

  // layer 3: t = relu(A) @ W3 ; out = aggregate(t) + b3 ; sigmoid
  head_kernel<<<cdiv_ll(n, TPB), TPB, 0, stream>>>(Ab, dis, W3, b3, t, out, n);
  out_scatter_kernel<<<cdiv_ll(nE, TPB), TPB, 0, stream>>>(ei, t, dis, out, nE);
  sigmoid_kernel<<<cdiv_ll(n, TPB), TPB, 0, stream>>>(out, n);
}